// Attention_45140106281189
// MI455X (gfx1250) — compile-verified
//
#include <hip/hip_runtime.h>
#include <math.h>

typedef __attribute__((ext_vector_type(2))) float v2f;
typedef __attribute__((ext_vector_type(8))) float v8f;

// ---------------------------------------------------------------------------
// Generic fp32 WMMA GEMM:  D[M,N] = A[M,K] @ B[K,N] (+bias)
// Block = 256 threads = 8 waves arranged 4(M) x 2(N); block tile 128 x 64.
// Each wave owns a 32x32 output block = 2x2 WMMA tiles -> 4 wmma per k-step
// with only 2 A-fragments + 2 B-fragments loaded (2x register reuse).
// AMODE (compile-time):
//   0: affine  A[m*lda + k*inca]         (AUNIT: inca==1 -> b64 vector loads)
//   1: tok2 reinterpret of x (1,C,H,W):  A[(m>>8)*131072 + (m&255) + k*256]
//   2: implicit im2col for sr 4x4/s4 conv on x:
//      k=(i,ky,kx), m=(oy,ox): x[i*65536 + (oy*4+ky)*256 + ox*4+kx]
// ---------------------------------------------------------------------------
template<int AMODE, bool AUNIT, bool HASBIAS>
__global__ __launch_bounds__(256)
void wmma_gemm_f32(const float* __restrict__ A, const float* __restrict__ B,
                   const float* __restrict__ bias, float* __restrict__ D,
                   int K, long lda, long inca, long ldb, long incb, long ldc)
{
  const int wave = threadIdx.x >> 5;
  const int lane = threadIdx.x & 31;
  const int mOff = blockIdx.y * 128 + (wave >> 1) * 32;
  const int nOff = blockIdx.x * 64 + (wave & 1) * 32;
  const int kk = (lane >> 4) << 1;          // lanes 0-15: K=0,1 ; lanes 16-31: K=2,3
  const int mA0 = mOff + (lane & 15), mA1 = mA0 + 16;
  const int nB0 = nOff + (lane & 15), nB1 = nB0 + 16;

  // Incremental A addressing (compile-time selected, no branches in loop).
  long ai0 = 0, ai1 = 0, aoff = 0, astep = 0, ab0 = 0, ab1 = 0;
  if constexpr (AMODE == 0) {
    ai0 = (long)mA0 * lda + (long)kk * inca;
    ai1 = (long)mA1 * lda + (long)kk * inca;
    aoff = inca; astep = 4 * inca;
  } else if constexpr (AMODE == 1) {
    ai0 = ((long)(mA0 >> 8)) * 131072L + (long)(mA0 & 255) + (long)kk * 256L;
    ai1 = ((long)(mA1 >> 8)) * 131072L + (long)(mA1 & 255) + (long)kk * 256L;
    aoff = 256; astep = 1024;
  } else {
    ab0 = ((long)(mA0 >> 6)) * 1024L + (long)((mA0 & 63) * 4) + (long)kk;
    ab1 = ((long)(mA1 >> 6)) * 1024L + (long)((mA1 & 63) * 4) + (long)kk;
  }
  long bi0 = (long)kk * ldb + (long)nB0 * incb;
  long bi1 = (long)kk * ldb + (long)nB1 * incb;
  const long bstep = 4 * ldb;

  v8f acc00 = {}, acc01 = {}, acc10 = {}, acc11 = {};
  #pragma unroll 4
  for (int k0 = 0; k0 < K; k0 += 4) {
    v2f a0, a1, b0, b1;
    if constexpr (AMODE == 2) {
      const int k = k0 + kk;
      const long kpart = ((long)(k >> 4)) * 65536L + (long)(((k >> 2) & 3) * 256);
      a0.x = A[kpart + ab0]; a0.y = A[kpart + ab0 + 1];
      a1.x = A[kpart + ab1]; a1.y = A[kpart + ab1 + 1];
    } else if constexpr (AUNIT) {
      a0 = *(const v2f*)(A + ai0);
      a1 = *(const v2f*)(A + ai1);
      ai0 += astep; ai1 += astep;
    } else {
      a0.x = A[ai0]; a0.y = A[ai0 + aoff];
      a1.x = A[ai1]; a1.y = A[ai1 + aoff];
      ai0 += astep; ai1 += astep;
    }
    b0.x = B[bi0]; b0.y = B[bi0 + ldb];
    b1.x = B[bi1]; b1.y = B[bi1 + ldb];
    __builtin_prefetch(B + bi0 + 8 * ldb, 0, 0);    // speculative, safe OOB
    bi0 += bstep; bi1 += bstep;
    acc00 = __builtin_amdgcn_wmma_f32_16x16x4_f32(false, a0, false, b0, (short)0, acc00, false, false);
    acc01 = __builtin_amdgcn_wmma_f32_16x16x4_f32(false, a0, false, b1, (short)0, acc01, false, false);
    acc10 = __builtin_amdgcn_wmma_f32_16x16x4_f32(false, a1, false, b0, (short)0, acc10, false, false);
    acc11 = __builtin_amdgcn_wmma_f32_16x16x4_f32(false, a1, false, b1, (short)0, acc11, false, false);
  }
  float bv0 = 0.0f, bv1 = 0.0f;
  if constexpr (HASBIAS) { bv0 = bias[nB0]; bv1 = bias[nB1]; }
  const int rb = mOff + ((lane >> 4) << 3);
  #pragma unroll
  for (int i = 0; i < 8; ++i) {
    D[(long)(rb + i)      * ldc + nB0] = acc00[i] + bv0;
    D[(long)(rb + i)      * ldc + nB1] = acc01[i] + bv1;
    D[(long)(rb + 16 + i) * ldc + nB0] = acc10[i] + bv0;
    D[(long)(rb + 16 + i) * ldc + nB1] = acc11[i] + bv1;
  }
}

// ---------------------------------------------------------------------------
// attn1: per (image-row b, head h). Q:256x64; K/V are the 64 SR tokens tiled 4x.
// K,V (16 KB each) staged once in LDS; 64-row score chunks in a 64 KB LDS tile.
// Total LDS = 96 KB (of 320 KB per WGP).
// ---------------------------------------------------------------------------
__global__ __launch_bounds__(256)
void attn1_kernel(const float* __restrict__ q1buf, const float* __restrict__ kv1,
                  float* __restrict__ x1buf, float* __restrict__ gmask)
{
  extern __shared__ float smem[];
  float* S  = smem;            // 64*256
  float* Kt = smem + 16384;    // 64*64
  float* Vt = Kt + 4096;       // 64*64
  const int b = blockIdx.x, h = blockIdx.y;
  const int wave = threadIdx.x >> 5, lane = threadIdx.x & 31;
  const int bs = (b & 63) * 64;             // tiled-key token base
  const int kk = (lane >> 4) << 1;

  // stage K and V tiles (64 tokens x 64 dims) into LDS
  for (int i = threadIdx.x; i < 4096; i += 256) {
    const int j = i >> 6, d = i & 63;
    const long base = (long)(bs + j) * 512 + h * 64 + d;
    Kt[i] = kv1[base];
    Vt[i] = kv1[base + 256];
  }
  __syncthreads();

  for (int c0 = 0; c0 < 256; c0 += 64) {
    // ---- S = scale * Q[c0..c0+63] @ K^T  (4x16 tiles, 8 waves) ----
    for (int t = wave; t < 64; t += 8) {
      const int mt = (t >> 4) << 4, nt = (t & 15) << 4;
      const int mA = mt + (lane & 15), nB = nt + (lane & 15);
      const float* Qrow = q1buf + ((long)b * 256 + (c0 + mA)) * 256 + h * 64 + kk;
      const float* Krow = Kt + (nB & 63) * 64 + kk;
      v8f acc = {};
      #pragma unroll 4
      for (int k0 = 0; k0 < 64; k0 += 4) {
        v2f a  = *(const v2f*)(Qrow + k0);
        v2f bb = *(const v2f*)(Krow + k0);
        acc = __builtin_amdgcn_wmma_f32_16x16x4_f32(false, a, false, bb, (short)0, acc, false, false);
      }
      const int rb = mt + ((lane >> 4) << 3);
      #pragma unroll
      for (int i = 0; i < 8; ++i) S[(rb + i) * 256 + nB] = acc[i] * 0.125f;
    }
    __syncthreads();
    // ---- row softmax (64 rows) ----
    if (threadIdx.x < 64) {
      float* row = S + threadIdx.x * 256;
      float mx = -3.4e38f;
      for (int j = 0; j < 256; ++j) mx = fmaxf(mx, row[j]);
      float sum = 0.f;
      for (int j = 0; j < 256; ++j) { float e = expf(row[j] - mx); row[j] = e; sum += e; }
      float inv = 1.0f / sum;
      for (int j = 0; j < 256; ++j) row[j] *= inv;
    }
    __syncthreads();
    // ---- gmask column sums (mean over 4 heads * 256 q) ----
    {
      float s = 0.f;
      const int k = threadIdx.x;
      for (int r = 0; r < 64; ++r) s += S[r * 256 + k];
      atomicAdd(&gmask[b * 256 + k], s * (1.0f / 1024.0f));
    }
    // ---- O = P @ V  (4x4 tiles, 8 waves; k split 4x64 so the tile wrap is free) ----
    for (int t = wave; t < 16; t += 8) {
      const int mt = (t >> 2) << 4, nt = (t & 3) << 4;
      const int mA = mt + (lane & 15), nB = nt + (lane & 15);
      v8f acc = {};
      #pragma unroll
      for (int rep = 0; rep < 4; ++rep) {
        const float* Pr = S + mA * 256 + rep * 64 + kk;
        const float* Vr = Vt + kk * 64 + nB;
        #pragma unroll 4
        for (int k0 = 0; k0 < 64; k0 += 4) {
          v2f a = *(const v2f*)(Pr + k0);
          v2f bb;
          bb.x = Vr[k0 * 64];
          bb.y = Vr[k0 * 64 + 64];
          acc = __builtin_amdgcn_wmma_f32_16x16x4_f32(false, a, false, bb, (short)0, acc, false, false);
        }
      }
      const int rb = mt + ((lane >> 4) << 3);
      #pragma unroll
      for (int i = 0; i < 8; ++i)
        x1buf[((long)b * 256 + (c0 + rb + i)) * 256 + h * 64 + nB] = acc[i];
    }
    __syncthreads();
  }
}

// ---------------------------------------------------------------------------
// attn2: windowed 8q x 8k x 64d attention. One wave per (window n, row r).
// Pure wave32 VALU + shuffles (problem far below a 16x16 WMMA tile).
// Handles the reference's flat-reinterpret of (Hc,H2,Wc,HD) as (H2,Hc,Wc,HD).
// ---------------------------------------------------------------------------
__global__ __launch_bounds__(256)
void attn2_kernel(const float* __restrict__ q2buf, const float* __restrict__ kv2,
                  float* __restrict__ x2buf, float* __restrict__ lmask)
{
  const int wave = threadIdx.x >> 5, lane = threadIdx.x & 31;
  const int p = blockIdx.x * 8 + wave;       // 0..32767 = (n, r)
  const int n = p >> 3, r = p & 7;
  const int a = n >> 10;                     // reinterpret "head" group
  const int wy = (n >> 5) & 31, wx = n & 31;
  const int Y4 = a * 256 + wy * 8 + r;
  const int y = Y4 >> 2, hh = Y4 & 3;        // actual (row, head) in the buffers

  const int k = lane & 7, q0 = lane >> 3;    // lane owns (q0,k) and (q0+4,k)
  const float4* Kv = (const float4*)(kv2 + (long)(y * 256 + wx * 8 + k) * 512 + hh * 64);
  const float4* Q0 = (const float4*)(q2buf + (long)(y * 256 + wx * 8 + q0) * 256 + hh * 64);
  const float4* Q1 = (const float4*)(q2buf + (long)(y * 256 + wx * 8 + q0 + 4) * 256 + hh * 64);
  float s0 = 0.f, s1 = 0.f;
  #pragma unroll
  for (int d4 = 0; d4 < 16; ++d4) {
    float4 kv = Kv[d4], qa = Q0[d4], qb = Q1[d4];
    s0 += qa.x * kv.x + qa.y * kv.y + qa.z * kv.z + qa.w * kv.w;
    s1 += qb.x * kv.x + qb.y * kv.y + qb.z * kv.z + qb.w * kv.w;
  }
  s0 *= 0.125f; s1 *= 0.125f;
  // softmax over k within 8-lane groups (two q's per lane, independent)
  float m0 = s0, m1 = s1;
  for (int mk = 1; mk < 8; mk <<= 1) {
    m0 = fmaxf(m0, __shfl_xor(m0, mk, 32));
    m1 = fmaxf(m1, __shfl_xor(m1, mk, 32));
  }
  float e0 = expf(s0 - m0), e1 = expf(s1 - m1);
  float t0 = e0, t1 = e1;
  for (int mk = 1; mk < 8; mk <<= 1) {
    t0 += __shfl_xor(t0, mk, 32);
    t1 += __shfl_xor(t1, mk, 32);
  }
  const float p0 = e0 / t0, p1 = e1 / t1;
  // lmask: mean over (h2=4, q=8); lanes l, l^8, l^16, l^24 share k
  float ls = p0 + p1;
  ls += __shfl_xor(ls, 8, 32);
  ls += __shfl_xor(ls, 16, 32);
  if (lane < 8)
    atomicAdd(&lmask[(wy * 8 + r) * 256 + wx * 8 + k], ls * (1.0f / 32.0f));
  // O = P @ V : lane owns output columns d0=lane, d1=lane+32
  const int d0 = lane, d1 = lane + 32;
  float vv0[8], vv1[8];
  #pragma unroll
  for (int j = 0; j < 8; ++j) {
    long vb = (long)(y * 256 + wx * 8 + j) * 512 + 256 + hh * 64;
    vv0[j] = kv2[vb + d0];
    vv1[j] = kv2[vb + d1];
  }
  #pragma unroll
  for (int q = 0; q < 8; ++q) {
    float o0 = 0.f, o1 = 0.f;
    #pragma unroll
    for (int j = 0; j < 8; ++j) {
      int src = (q & 3) * 8 + j;
      float pv = (q < 4) ? __shfl(p0, src, 32) : __shfl(p1, src, 32);
      o0 += pv * vv0[j];
      o1 += pv * vv1[j];
    }
    long ob = (long)((wy * 8 + r) * 256 + wx * 8 + q) * 256 + a * 64;
    x2buf[ob + d0] = o0;
    x2buf[ob + d1] = o1;
  }
}

// depthwise 3x3, pad 1; lin is token-major (HW,512); out is channel-major (C,H,W)
__global__ __launch_bounds__(256)
void dwconv3x3(const float* __restrict__ lin, const float* __restrict__ w,
               const float* __restrict__ bias, float* __restrict__ out)
{
  const int x = threadIdx.x, y = blockIdx.x, c = blockIdx.y;
  float acc = bias[c];
  for (int dy = -1; dy <= 1; ++dy) {
    int yy = y + dy;
    if (yy < 0 || yy >= 256) continue;
    for (int dx = -1; dx <= 1; ++dx) {
      int xx = x + dx;
      if (xx < 0 || xx >= 256) continue;
      acc += lin[((long)yy * 256 + xx) * 512 + c] * w[c * 9 + (dy + 1) * 3 + (dx + 1)];
    }
  }
  out[(long)c * 65536 + y * 256 + x] = acc;
}

__global__ __launch_bounds__(256)
void chan_stats(const float* __restrict__ xs, float* __restrict__ chsum, float* __restrict__ chsq)
{
  __shared__ float ssum[256], ssq[256];
  const int c = blockIdx.x;
  float s = 0.f, q = 0.f;
  for (int t = threadIdx.x; t < 4096; t += 256) {
    float v = xs[(long)t * 512 + c];
    s += v; q += v * v;
  }
  ssum[threadIdx.x] = s; ssq[threadIdx.x] = q;
  __syncthreads();
  for (int st = 128; st > 0; st >>= 1) {
    if (threadIdx.x < st) {
      ssum[threadIdx.x] += ssum[threadIdx.x + st];
      ssq[threadIdx.x]  += ssq[threadIdx.x + st];
    }
    __syncthreads();
  }
  if (threadIdx.x == 0) { chsum[c] = ssum[0]; chsq[c] = ssq[0]; }
}

__global__ __launch_bounds__(256)
void global_stats(const float* __restrict__ chsum, const float* __restrict__ chsq,
                  float* __restrict__ g)
{
  __shared__ float ssum[256], ssq[256];
  float s = 0.f, q = 0.f;
  for (int c = threadIdx.x; c < 512; c += 256) { s += chsum[c]; q += chsq[c]; }
  ssum[threadIdx.x] = s; ssq[threadIdx.x] = q;
  __syncthreads();
  for (int st = 128; st > 0; st >>= 1) {
    if (threadIdx.x < st) {
      ssum[threadIdx.x] += ssum[threadIdx.x + st];
      ssq[threadIdx.x]  += ssq[threadIdx.x + st];
    }
    __syncthreads();
  }
  if (threadIdx.x == 0) { g[0] = ssum[0]; g[1] = ssq[0]; }
}

// AdaILN (unbiased var) + softmax(rho) blend + exact GELU, in place on xs (4096,512)
__global__ __launch_bounds__(256)
void adailn_gelu(float* __restrict__ xs, const float* __restrict__ chsum,
                 const float* __restrict__ chsq, const float* __restrict__ g,
                 const float* __restrict__ rho, const float* __restrict__ gamma,
                 const float* __restrict__ beta)
{
  const long e = (long)blockIdx.x * 256 + threadIdx.x;
  const int c = (int)(e & 511);
  float v = xs[e];
  const float nIn = 4096.0f, nLn = 2097152.0f;
  float im = chsum[c] / nIn;
  float iv = (chsq[c] - nIn * im * im) / (nIn - 1.0f);
  float lm = g[0] / nLn;
  float lv = (g[1] - nLn * lm * lm) / (nLn - 1.0f);
  float r0e = expf(rho[c * 2]), r1e = expf(rho[c * 2 + 1]);
  float rs = r0e + r1e;
  float o = (r0e / rs) * (v - im) * rsqrtf(iv + 1e-5f)
          + (r1e / rs) * (v - lm) * rsqrtf(lv + 1e-5f);
  o = o * gamma[c] + beta[c];
  xs[e] = 0.5f * o * (1.0f + erff(o * 0.70710678118654752f));
}

// Ein[t][c] = concat(x1,x2)[t][c] + lepe-reinterpret[t][c]
__global__ __launch_bounds__(256)
void ein_build(const float* __restrict__ x1buf, const float* __restrict__ x2buf,
               const float* __restrict__ conv, float* __restrict__ Ein)
{
  const long e = (long)blockIdx.x * 256 + threadIdx.x;
  const int c = (int)(e & 511);
  const long t = e >> 9;
  const int y = (int)(t >> 8), x = (int)(t & 255);
  float part;
  if (c < 256) part = x1buf[t * 256 + c];
  else         part = x2buf[(long)(x * 256 + (c - 256)) * 256 + y];
  float lep = conv[(long)(2 * y + (c >> 8)) * 65536 + (c & 255) * 256 + x];
  Ein[e] = part + lep;
}

__global__ __launch_bounds__(256)
void mask_kernel(const float* __restrict__ gmask, const float* __restrict__ lmask,
                 float* __restrict__ m1, float* __restrict__ m2)
{
  int i = blockIdx.x * 256 + threadIdx.x;   // < 65536
  int y = i >> 8, x = i & 255;
  float m = lmask[i] + gmask[i];
  m1[i] = m;
  m2[x * 256 + y] = m;
}

extern "C" void kernel_launch(void* const* d_in, const int* in_sizes, int n_in,
                              void* d_out, int out_size, void* d_ws, size_t ws_size,
                              hipStream_t stream)
{
  const float* x           = (const float*)d_in[0];
  const float* q1_w        = (const float*)d_in[1];
  const float* kv1_w       = (const float*)d_in[2];
  const float* q2_w        = (const float*)d_in[3];
  const float* kv2_w       = (const float*)d_in[4];
  const float* sr_w        = (const float*)d_in[5];
  const float* sr_b        = (const float*)d_in[6];
  const float* rho         = (const float*)d_in[7];
  const float* gamma       = (const float*)d_in[8];
  const float* beta        = (const float*)d_in[9];
  const float* lepe_lin_w  = (const float*)d_in[10];
  const float* lepe_lin_b  = (const float*)d_in[11];
  const float* lepe_conv_w = (const float*)d_in[12];
  const float* lepe_conv_b = (const float*)d_in[13];
  const float* proj_w      = (const float*)d_in[14];
  const float* proj_b      = (const float*)d_in[15];

  float* ws    = (float*)d_ws;
  float* lin   = ws;                    // 33554432 (reused as Ein later)
  float* conv  = ws + 33554432L;        // 33554432
  float* q1b   = ws + 67108864L;        // 16777216
  float* q2b   = ws + 83886080L;        // 16777216
  float* kv2b  = ws + 100663296L;       // 33554432
  float* x1b   = ws + 134217728L;       // 16777216
  float* x2b   = ws + 150994944L;       // 16777216
  float* xs    = ws + 167772160L;       // 2097152
  float* kv1b  = ws + 169869312L;       // 2097152
  float* chsum = ws + 171966464L;       // 512
  float* chsq  = chsum + 512;           // 512
  float* gst   = chsq + 512;            // 2
  float* gmask = gst + 2;               // 65536
  float* lmask = gmask + 65536;         // 65536

  hipMemsetAsync(gmask, 0, 65536 * sizeof(float), stream);
  hipMemsetAsync(lmask, 0, 65536 * sizeof(float), stream);

  dim3 blk(256);
  // lepe linear: xt(65536x512) @ W(512x512) + b  (A = x via lda=1, inca=65536)
  wmma_gemm_f32<0, false, true><<<dim3(8, 512), blk, 0, stream>>>(
      x, lepe_lin_w, lepe_lin_b, lin, 512, 1L, 65536L, 512L, 1L, 512L);
  dwconv3x3<<<dim3(256, 512), blk, 0, stream>>>(lin, lepe_conv_w, lepe_conv_b, conv);
  // q1, q2: xt @ (512x256)
  wmma_gemm_f32<0, false, false><<<dim3(4, 512), blk, 0, stream>>>(
      x, q1_w, nullptr, q1b, 512, 1L, 65536L, 256L, 1L, 256L);
  wmma_gemm_f32<0, false, false><<<dim3(4, 512), blk, 0, stream>>>(
      x, q2_w, nullptr, q2b, 512, 1L, 65536L, 256L, 1L, 256L);
  // sr conv as implicit-im2col GEMM: (4096 x 8192) @ (8192 x 512); B = sr_w (OIHW -> n*8192+k)
  wmma_gemm_f32<2, false, true><<<dim3(8, 32), blk, 0, stream>>>(
      x, sr_w, sr_b, xs, 8192, 0L, 0L, 1L, 8192L, 512L);
  chan_stats<<<512, blk, 0, stream>>>(xs, chsum, chsq);
  global_stats<<<1, blk, 0, stream>>>(chsum, chsq, gst);
  adailn_gelu<<<8192, blk, 0, stream>>>(xs, chsum, chsq, gst, rho, gamma, beta);
  // kv1 = xs_norm(4096x512) @ kv1_w(512x512)   (A rows contiguous -> b64 loads)
  wmma_gemm_f32<0, true, false><<<dim3(8, 32), blk, 0, stream>>>(
      xs, kv1_w, nullptr, kv1b, 512, 512L, 1L, 512L, 1L, 512L);
  // kv2 = tok2(reinterpret of x) @ kv2_w(512x512)
  wmma_gemm_f32<1, false, false><<<dim3(8, 512), blk, 0, stream>>>(
      x, kv2_w, nullptr, kv2b, 512, 0L, 0L, 512L, 1L, 512L);
  // fused attentions (attn1: 96 KB dynamic LDS)
  attn1_kernel<<<dim3(256, 4), blk, (16384 + 8192) * sizeof(float), stream>>>(q1b, kv1b, x1b, gmask);
  attn2_kernel<<<4096, blk, 0, stream>>>(q2b, kv2b, x2b, lmask);
  // Ein = concat(x1,x2) + lepe  (written into lin's region, which is free now)
  ein_build<<<131072, blk, 0, stream>>>(x1b, x2b, conv, lin);
  // out = Ein @ proj_w + proj_b  -> d_out[0 : 65536*512]
  wmma_gemm_f32<0, true, true><<<dim3(8, 512), blk, 0, stream>>>(
      lin, proj_w, proj_b, (float*)d_out, 512, 512L, 1L, 512L, 1L, 512L);
  // masks
  float* m1 = (float*)d_out + 33554432L;
  mask_kernel<<<256, blk, 0, stream>>>(gmask, lmask, m1, m1 + 65536);
}